// GNNEncoder_48077863911694
// MI455X (gfx1250) — compile-verified
//
#include <hip/hip_runtime.h>
#include <cmath>

typedef float v2f __attribute__((ext_vector_type(2)));
typedef float v8f __attribute__((ext_vector_type(8)));

#define HID 128

// ---------- degree / normalization ----------
__global__ __launch_bounds__(256) void k_deg_init(float* __restrict__ deg, int n) {
    int i = blockIdx.x * 256 + threadIdx.x;
    if (i < n) deg[i] = 1.0f;  // self-loop contribution
}

__global__ __launch_bounds__(256) void k_deg_acc(const int* __restrict__ dst,
                                                 float* __restrict__ deg, int e) {
    int i = blockIdx.x * 256 + threadIdx.x;
    if (i < e) atomicAdd(&deg[dst[i]], 1.0f);
}

__global__ __launch_bounds__(256) void k_rsqrt(float* __restrict__ deg, int n) {
    int i = blockIdx.x * 256 + threadIdx.x;
    if (i < n) deg[i] = rsqrtf(deg[i]);  // deg >= 1 always (self-loops)
}

// ---------- GEMM: Out[M x 128] = A[M x 128] * W[128 x 128] (+bias) ----------
// fp32 WMMA 16x16x4, one wave per 16x16 output tile, 8 waves = 16x128 strip.
__global__ __launch_bounds__(256) void k_gemm_wmma(const float* __restrict__ A,
                                                   const float* __restrict__ W,
                                                   const float* __restrict__ bias,
                                                   float* __restrict__ Out,
                                                   int M, int useBias) {
    __shared__ float sA[16 * HID];  // 8 KB row strip
    const int tid = threadIdx.x;
    const int rowBase = blockIdx.x * 16;

    // cooperative stage of the 16x128 A strip (512 float4 / 256 threads)
    {
        float4* s4 = (float4*)sA;
        const float4* a4 = (const float4*)(A + (size_t)rowBase * HID);
        for (int i = tid; i < 512; i += 256) {
            int r = i >> 5;  // 32 float4 per row
            s4[i] = (rowBase + r < M) ? a4[i] : make_float4(0.f, 0.f, 0.f, 0.f);
        }
    }
    __syncthreads();

    const int wave = tid >> 5;           // col tile 0..7
    const int lane = tid & 31;
    const int row  = lane & 15;          // A row within tile / B,N index
    const int koff = (lane >> 4) << 1;   // lanes 16-31 hold K+2,K+3
    const int colBase = wave << 4;
    const int col = colBase + row;       // output / W column for this lane

    v8f acc = {};
    for (int k = 0; k < HID; k += 4) {
        v2f a = *(const v2f*)&sA[row * HID + k + koff];   // A[row][k+koff], [k+koff+1]
        v2f b;
        b.x = W[(k + koff)     * HID + col];              // B[k+koff][col]
        b.y = W[(k + koff + 1) * HID + col];
        acc = __builtin_amdgcn_wmma_f32_16x16x4_f32(
            /*neg_a=*/false, a, /*neg_b=*/false, b,
            /*c_mod=*/(short)0, acc, /*reuse_a=*/false, /*reuse_b=*/false);
    }

    const float bv = useBias ? bias[col] : 0.0f;
    const int mHalf = (lane >> 4) << 3;  // VGPR r holds M = r + 8*(lane>>4)
#pragma unroll
    for (int r = 0; r < 8; ++r) {
        int mm = rowBase + mHalf + r;
        if (mm < M) Out[(size_t)mm * HID + col] = acc[r] + bv;
    }
}

// ---------- agg init with self-loop: agg[i][f] = inv[i]^2 * m[i][f] ----------
__global__ __launch_bounds__(256) void k_selfloop(const float* __restrict__ inv,
                                                  const float* __restrict__ m,
                                                  float* __restrict__ agg, int total) {
    int i = blockIdx.x * 256 + threadIdx.x;
    if (i < total) {
        float s = inv[i >> 7];  // HID == 128
        agg[i] = s * s * m[i];
    }
}

// ---------- edge scatter: one wave per edge, float4 per lane ----------
__global__ __launch_bounds__(256) void k_scatter(const int* __restrict__ src,
                                                 const int* __restrict__ dst,
                                                 const float* __restrict__ inv,
                                                 const float* __restrict__ m,
                                                 float* __restrict__ agg, int e) {
    int gid  = blockIdx.x * 256 + threadIdx.x;
    int edge = gid >> 5;
    int lane = gid & 31;
    if (edge >= e) return;
    int s = src[edge];
    int d = dst[edge];
    float c = inv[s] * inv[d];
    float4 v = ((const float4*)(m + (size_t)s * HID))[lane];
    float* arow = agg + (size_t)d * HID + lane * 4;
    atomicAdd(arow + 0, c * v.x);
    atomicAdd(arow + 1, c * v.y);
    atomicAdd(arow + 2, c * v.z);
    atomicAdd(arow + 3, c * v.w);
}

// ---------- h = gelu_exact(agg + b) + h ----------
__global__ __launch_bounds__(256) void k_gelu_res(const float* __restrict__ agg,
                                                  const float* __restrict__ b,
                                                  float* __restrict__ h, int total) {
    int i = blockIdx.x * 256 + threadIdx.x;
    if (i < total) {
        float z = agg[i] + b[i & (HID - 1)];
        float g = 0.5f * z * (1.0f + erff(z * 0.70710678118654752440f));
        h[i] = g + h[i];
    }
}

extern "C" void kernel_launch(void* const* d_in, const int* in_sizes, int n_in,
                              void* d_out, int out_size, void* d_ws, size_t ws_size,
                              hipStream_t stream) {
    const float* x        = (const float*)d_in[0];
    const int*   eidx     = (const int*)d_in[1];
    const float* W_in     = (const float*)d_in[2];
    const float* b_in     = (const float*)d_in[3];
    const float* W_layers = (const float*)d_in[4];
    const float* b_layers = (const float*)d_in[5];

    const int N = in_sizes[0] / HID;   // 50000
    const int E = in_sizes[1] / 2;     // 600000
    const int nLayers = in_sizes[4] / (HID * HID);  // 3

    const int* srcIdx = eidx;          // edge_index[0]
    const int* dstIdx = eidx + E;      // edge_index[1]

    float* h = (float*)d_out;

    // workspace layout: inv[N] | m[N*HID] | agg[N*HID]
    char* ws = (char*)d_ws;
    size_t invBytes = ((size_t)N * sizeof(float) + 255) & ~(size_t)255;
    float* inv = (float*)ws;
    float* m   = (float*)(ws + invBytes);
    float* agg = m + (size_t)N * HID;

    const int total = N * HID;
    const int ewBlocks   = (E * 32 + 255) / 256;   // wave-per-edge kernels
    const int elemBlocks = (total + 255) / 256;
    const int nBlocks    = (N + 255) / 256;
    const int eBlocks    = (E + 255) / 256;
    const int gemmBlocks = (N + 15) / 16;

    // normalization coefficients
    k_deg_init<<<nBlocks, 256, 0, stream>>>(inv, N);
    k_deg_acc <<<eBlocks, 256, 0, stream>>>(dstIdx, inv, E);
    k_rsqrt   <<<nBlocks, 256, 0, stream>>>(inv, N);

    // input projection: h = x @ W_in + b_in
    k_gemm_wmma<<<gemmBlocks, 256, 0, stream>>>(x, W_in, b_in, h, N, 1);

    for (int l = 0; l < nLayers; ++l) {
        const float* Wl = W_layers + (size_t)l * HID * HID;
        const float* bl = b_layers + (size_t)l * HID;
        // m = h @ W_l
        k_gemm_wmma<<<gemmBlocks, 256, 0, stream>>>(h, Wl, nullptr, m, N, 0);
        // agg = self-loop term
        k_selfloop<<<elemBlocks, 256, 0, stream>>>(inv, m, agg, total);
        // agg += scatter of normalized messages
        k_scatter<<<ewBlocks, 256, 0, stream>>>(srcIdx, dstIdx, inv, m, agg, E);
        // h = gelu(agg + b_l) + h
        k_gelu_res<<<elemBlocks, 256, 0, stream>>>(agg, bl, h, total);
    }
}